// BatteryRNNCell_82523501626118
// MI455X (gfx1250) — compile-verified
//
#include <hip/hip_runtime.h>

// ---------------------------------------------------------------------------
// BatteryRNNCell on MI455X (gfx1250): elementwise, bandwidth-bound.
// ~159 MB traffic/call -> ~6.8 us HBM floor (working set < 192 MB L2).
// VALU minimization (to stay under the memory roofline):
//   - fp32 divisions -> v_rcp_f32 / v_rsq_f32
//   - _ve_sum -> A0*m + m^2*P(m) - tx*Q(m), two compile-time Horner chains
//   - Vep-Ven log terms merged into ONE v_log_f32:
//       log((1-xp)/xp) - log((1-xn)/xn) = log(((1-xp)*xn)/(xp*(1-xn)))
//   - bare v_log_f32 (base-2) with ln2 folded into the Tb scale constants
//   - bare v_sqrt_f32 via __builtin_amdgcn_sqrtf
//   - pointer-bumped grid-stride loop
// WMMA intentionally unused: no matrix structure exists in this workload
// (the only contraction is length-13 vs compile-time scalars = 22 FMAs).
// gfx1250-specific path: global_prefetch_b8 for the next grid-stride chunk.
// ---------------------------------------------------------------------------

#define F_CONST      96487.0
#define LN2          0.6931471805599453
#define INV_F        ((float)(1.0 / F_CONST))
// (R/(F*alpha))*ln2 with alpha=0.5 -> multiplies log2 in asinh terms
#define CT_L         ((float)(2.0 * 8.3144621 / F_CONST * LN2))
// (R/F)*ln2 -> multiplies log2 of the merged electrode ratio
#define RTF_L        ((float)(8.3144621 / F_CONST * LN2))
#define INV_SN       ((float)(1.0 / 0.000437545))
#define INV_SP       ((float)(1.0 / 0.00030962))
#define HALF_INV_KN  ((float)(0.5 / 2120.96))              // 1/(2*KN)
#define HALF_INV_KP  ((float)(0.5 / 248898.0))             // 1/(2*KP)
#define RO_C         0.117215f
#define INV_TDIFF    ((float)(1.0 / 7000000.0))
#define INV_TO       ((float)(1.0 / 6.08671))
#define INV_TSN      ((float)(1.0 / 1001.38))
#define INV_TSP      ((float)(1.0 / 46.4311))
#define U0_DIFF      ((float)(4.03 - 0.01))                // U0P - U0N
#define INV_VOLB     ((float)(1.0 / 1.8e-05))              // VOL_B = 0.9*2e-5
#define INV_VOLS     ((float)(1.0 / 2.0e-06))              // VOL_S = 0.1*2e-5
#define INV_QSMAX    ((float)(0.6 / 7600.0 / 0.1))         // 1/(QMAX*VOL_S/VOL)
#define AN0_OVER_F   ((float)(86.19 / F_CONST))            // AN: only A[0] != 0

// asinh(z) in base-2: log2(z + sqrt(z^2+1)); caller multiplies by scale*ln2
__device__ __forceinline__ float asinh_log2(float z) {
    return __builtin_amdgcn_logf(z + __builtin_amdgcn_sqrtf(fmaf(z, z, 1.0f)));
}

__device__ __forceinline__ float ve_sum_p(float x) {
    // sum_{k=0..12} AP[k]*m^(k-1)*(m^2 - tx*k)/F  (k=0 term: AP0*m),
    // m=2x-1, tx=2x(1-x).  Rewritten: (AP0*m + m^2*P(m) - tx*Q(m))/F with
    // P(m)=sum_{k>=1} AP[k] m^{k-1},  Q(m)=sum_{k>=1} k*AP[k] m^{k-1}.
    constexpr float AP[13] = {
        -31593.7f, 0.106747f, 24606.4f, -78561.9f, 13317.9f, 307387.0f,
        84916.1f, -1074690.0f, 2285.04f, 990894.0f, 283920.0f, -161513.0f,
        -469218.0f};
    const float m  = fmaf(2.0f, x, -1.0f);
    const float tx = 2.0f * x * (1.0f - x);
    float P = AP[12];
    float Q = 12.0f * AP[12];
#pragma unroll
    for (int k = 11; k >= 1; --k) {          // fully unrolled; (float)k*AP[k]
        P = fmaf(P, m, AP[k]);               // folds to literals -> two
        Q = fmaf(Q, m, (float)k * AP[k]);    // independent 11-FMA chains
    }
    const float s = fmaf(m * m, P, fmaf(-tx, Q, AP[0] * m));
    return s * INV_F;
}

__global__ __launch_bounds__(256) void battery_cell_kernel(
    const float* __restrict__ inp,   // (B,)
    const float* __restrict__ st,    // (B,8) row-major
    float* __restrict__ z_out,       // (B,2)
    float* __restrict__ x_out,       // (B,8)
    int n)
{
    const int stride = gridDim.x * blockDim.x;
    int t = blockIdx.x * blockDim.x + threadIdx.x;

    // pointer-bumped bases (no per-iteration 64-bit address rebuild)
    const float*  ip = inp + t;
    const float4* sp = reinterpret_cast<const float4*>(st) + (size_t)t * 2;
    float2*       zp = reinterpret_cast<float2*>(z_out) + t;
    float4*       xp = reinterpret_cast<float4*>(x_out) + (size_t)t * 2;

    for (; t < n; t += stride) {
        // gfx1250 prefetch (global_prefetch_b8): next chunk's state row+input
        if (t + stride < n) {
            __builtin_prefetch(sp + 2 * (size_t)stride, 0, 1);
            __builtin_prefetch(ip + stride, 0, 1);
        }

        const float  cur = *ip;
        const float4 s0  = sp[0];
        const float4 s1  = sp[1];
        const float Tb  = s0.x, Vo  = s0.y, Vsn = s0.z, Vsp = s0.w;
        const float qnB = s1.x, qnS = s1.y, qpB = s1.z, qpS = s1.w;

        // surface stoichiometries and bulk<->surface diffusion
        const float xnS  = qnS * INV_QSMAX;
        const float xpS  = qpS * INV_QSMAX;
        const float qdDn = (qnB * INV_VOLB - qnS * INV_VOLS) * INV_TDIFF;
        const float qdDp = (qpB * INV_VOLB - qpS * INV_VOLS) * INV_TDIFF;

        // Jn/(2*Jn0) = Jn * (1/(2*KN)) * rsq(x*(1-x))   (alpha = 0.5)
        const float Jn = cur * INV_SN;
        const float Jp = cur * INV_SP;
        const float an = Jn * HALF_INV_KN * __builtin_amdgcn_rsqf(xnS * (1.0f - xnS));
        const float ap = Jp * HALF_INV_KP * __builtin_amdgcn_rsqf(xpS * (1.0f - xpS));

        const float cTl    = CT_L * Tb;      // (R*Tb*ln2)/(F*alpha)
        const float VsnNom = cTl * asinh_log2(an);
        const float VspNom = cTl * asinh_log2(ap);

        // state update (DT = 1)
        const float Vo_n  = fmaf(fmaf(cur, RO_C, -Vo), INV_TO,  Vo);
        const float Vsn_n = fmaf(VsnNom - Vsn,        INV_TSN, Vsn);
        const float Vsp_n = fmaf(VspNom - Vsp,        INV_TSP, Vsp);
        const float qnB_n = qnB - qdDn;
        const float qnS_n = qnS + (qdDn - cur);
        const float qpB_n = qpB - qdDp;
        const float qpS_n = qpS + (cur + qdDp);

        // V = Vep - Ven - Vo_n - Vsn_n - Vsp_n with the two electrode logs
        // merged: rtF*log( ((1-xp)*xn) / (xp*(1-xn)) ), one v_log + one v_rcp
        const float xn2 = qnS_n * INV_QSMAX;
        const float xp2 = qpS_n * INV_QSMAX;
        const float num   = (1.0f - xp2) * xn2;
        const float den   = xp2 * (1.0f - xn2);
        const float ratio = num * __builtin_amdgcn_rcpf(den);
        const float elog  = (RTF_L * Tb) * __builtin_amdgcn_logf(ratio);
        const float V = U0_DIFF + elog + ve_sum_p(xp2)
                      - AN0_OVER_F * fmaf(2.0f, xn2, -1.0f)
                      - Vo_n - Vsn_n - Vsp_n;

        // outputs: Z (B,2) then X_next (B,8), coalesced vector stores
        *zp   = make_float2(Tb - 273.15f, V);
        xp[0] = make_float4(Tb, Vo_n, Vsn_n, Vsp_n);
        xp[1] = make_float4(qnB_n, qnS_n, qpB_n, qpS_n);

        ip += stride;
        sp += 2 * (size_t)stride;
        zp += stride;
        xp += 2 * (size_t)stride;
    }
}

extern "C" void kernel_launch(void* const* d_in, const int* in_sizes, int n_in,
                              void* d_out, int out_size, void* d_ws, size_t ws_size,
                              hipStream_t stream) {
    (void)n_in; (void)out_size; (void)d_ws; (void)ws_size;
    const float* inp = reinterpret_cast<const float*>(d_in[0]); // inputs (B,1)
    const float* st  = reinterpret_cast<const float*>(d_in[1]); // states (B,8)
    const int n = in_sizes[0];                                  // B

    float* z = reinterpret_cast<float*>(d_out);                 // (B,2)
    float* x = reinterpret_cast<float*>(d_out) + (size_t)2 * n; // (B,8)

    const int block = 256;                    // 8 wave32 per block
    int grid = (n + block - 1) / block;
    if (grid > 2048) grid = 2048;             // 4 elements/thread at B=2^21
    if (grid < 1) grid = 1;

    battery_cell_kernel<<<grid, block, 0, stream>>>(inp, st, z, x, n);
}